// Laplacian3D_31602369364248
// MI455X (gfx1250) — compile-verified
//
#include <hip/hip_runtime.h>
#include <hip/hip_bf16.h>

// Laplacian3D, dilations {1,2,4}, coeff = 0.25*sigmoid(diff_params) per channel.
// x: (2,64,64,64,64) fp32, diff_params: (3,64) fp32. Bandwidth-bound:
// ~256 MiB HBM traffic -> ~11us floor at 23.3 TB/s. Strategy: async
// global->LDS halo staging (gfx1250 ASYNCcnt path) + LDS-served 19-tap stencil.

#define NB 2
#define NC 64
#define ND 64
#define NH 64
#define NW 64
#define DT 8
#define HT 8
#define HALO 4
#define LDS_D (DT + 2 * HALO) // 16
#define LDS_H (HT + 2 * HALO) // 16

#define AS1 __attribute__((address_space(1)))
#define AS3 __attribute__((address_space(3)))

typedef int v4i __attribute__((ext_vector_type(4)));

static __device__ __forceinline__ void async_copy_b128(const float* gp, float* lp) {
#if __has_builtin(__builtin_amdgcn_global_load_async_to_lds_b128)
  __builtin_amdgcn_global_load_async_to_lds_b128((AS1 v4i*)gp, (AS3 v4i*)lp, 0, 0);
#else
  unsigned lds_addr = (unsigned)(unsigned long long)(AS3 void*)lp;
  asm volatile("global_load_async_to_lds_b128 %0, %1, off"
               :: "v"(lds_addr), "v"(gp)
               : "memory");
#endif
}

static __device__ __forceinline__ void wait_asynccnt0() {
#if __has_builtin(__builtin_amdgcn_s_wait_asynccnt)
  __builtin_amdgcn_s_wait_asynccnt(0);
#else
  asm volatile("s_wait_asynccnt 0" ::: "memory");
#endif
}

extern "C" __global__ __launch_bounds__(256)
void lap3d_dilated_kernel(const float* __restrict__ x,
                          const float* __restrict__ dp,
                          float* __restrict__ out) {
  __shared__ float tile[LDS_D][LDS_H][NW]; // 64 KiB

  const int tid = threadIdx.x;
  const int blk = blockIdx.x;
  const int yt = blk & 7;
  const int zt = (blk >> 3) & 7;
  const int bc = blk >> 6;              // 0..127 = b*64 + c
  const int c  = bc & (NC - 1);
  const int z0 = zt * DT;
  const int y0 = yt * HT;

  const float* xs = x   + (size_t)bc * (ND * NH * NW);
  float*       os = out + (size_t)bc * (ND * NH * NW);

  // Per-channel coefficients: 0.25 * sigmoid(p)
  const float c1 = 0.25f / (1.0f + __expf(-dp[0 * NC + c]));
  const float c2 = 0.25f / (1.0f + __expf(-dp[1 * NC + c]));
  const float c4 = 0.25f / (1.0f + __expf(-dp[2 * NC + c]));
  const float cc = -6.0f * (c1 + c2 + c4);

  // ---- Cooperative halo load: 16x16 rows x 64 floats, b128 chunks ----
  // 4096 16-byte chunks / 256 threads = 16 per thread.
  #pragma unroll
  for (int pass = 0; pass < 16; ++pass) {
    const int q   = pass * 256 + tid;
    const int wo  = (q & 15) << 2;      // float offset within row, 16B aligned
    const int row = q >> 4;             // 0..255
    const int dy  = row & (LDS_H - 1);
    const int dz  = row >> 4;
    const int gz  = z0 - HALO + dz;
    const int gy  = y0 - HALO + dy;
    float* lp = &tile[dz][dy][wo];
    if ((unsigned)gz < (unsigned)ND && (unsigned)gy < (unsigned)NH) {
      const float* gp = xs + ((size_t)gz * NH + gy) * NW + wo;
      async_copy_b128(gp, lp);          // ASYNCcnt-tracked, bypasses VGPRs
    } else {
      *(float4*)lp = make_float4(0.f, 0.f, 0.f, 0.f); // zero halo (ds store)
    }
  }
  wait_asynccnt0();   // drain this wave's async LDS writes
  __syncthreads();    // waits DScnt too; all waves' tile data visible

  // ---- Stencil from LDS: 8x8x64 outputs, 16 points per thread ----
  for (int i = 0; i < 16; ++i) {
    const int p = i * 256 + tid;
    const int w = p & (NW - 1);
    const int y = (p >> 6) & (HT - 1);
    const int z = p >> 9;
    const int lz = z + HALO;
    const int ly = y + HALO;

    const float xc = tile[lz][ly][w];

    float s1 = tile[lz - 1][ly][w] + tile[lz + 1][ly][w]
             + tile[lz][ly - 1][w] + tile[lz][ly + 1][w];
    if (w >= 1)      s1 += tile[lz][ly][w - 1];
    if (w <= NW - 2) s1 += tile[lz][ly][w + 1];

    float s2 = tile[lz - 2][ly][w] + tile[lz + 2][ly][w]
             + tile[lz][ly - 2][w] + tile[lz][ly + 2][w];
    if (w >= 2)      s2 += tile[lz][ly][w - 2];
    if (w <= NW - 3) s2 += tile[lz][ly][w + 2];

    float s4 = tile[lz - 4][ly][w] + tile[lz + 4][ly][w]
             + tile[lz][ly - 4][w] + tile[lz][ly + 4][w];
    if (w >= 4)      s4 += tile[lz][ly][w - 4];
    if (w <= NW - 5) s4 += tile[lz][ly][w + 4];

    const float r = fmaf(cc, xc, fmaf(c1, s1, fmaf(c2, s2, c4 * s4)));
    os[((size_t)(z0 + z) * NH + (y0 + y)) * NW + w] = r;
  }
}

extern "C" void kernel_launch(void* const* d_in, const int* in_sizes, int n_in,
                              void* d_out, int out_size, void* d_ws, size_t ws_size,
                              hipStream_t stream) {
  const float* x  = (const float*)d_in[0];   // (2,64,64,64,64) fp32
  const float* dp = (const float*)d_in[1];   // (3,64) fp32
  float* out = (float*)d_out;                // same shape as x

  const int nblocks = NB * NC * (ND / DT) * (NH / HT); // 8192
  lap3d_dilated_kernel<<<dim3(nblocks), dim3(256), 0, stream>>>(x, dp, out);
}